// VariationalNetwork_36575941492926
// MI455X (gfx1250) — compile-verified
//
#include <hip/hip_runtime.h>
#include <math.h>

// ---------------------------------------------------------------------------
// VariationalNetwork on MI455X (gfx1250).
// FFT-256 as radix-16 => two complex 16x16 GEMMs => V_WMMA_F32_16X16X4_F32.
// Working set (~100MB) is L2-resident on MI455X (192MB L2), so we optimize
// for matrix-pipe FLOPs + LDS traffic, not HBM.
// ---------------------------------------------------------------------------

typedef float v2f __attribute__((ext_vector_type(2)));
typedef float v8f __attribute__((ext_vector_type(8)));

#define HW   65536
#define BT   96
#define NPIX ((size_t)BT * HW)   // 6291456

__device__ __forceinline__ v8f wmma4(v2f a, v2f b, v8f c) {
  // D = A(16x4) * B(4x16) + C(16x16), all f32. neg flags must be 0 for f32.
  return __builtin_amdgcn_wmma_f32_16x16x4_f32(false, a, false, b,
                                               (short)0, c, false, false);
}

// D16[m][k] = exp(-2*pi*i*m*k/16) = cos(.) - i*sin(.). D16 is symmetric, so
// the A-layout chunks (lane holds D[lo][k0],D[lo][k0+1]) are bit-identical to
// the B-layout chunks (lane holds D[k0][lo],D[k0+1][lo]).
__device__ __forceinline__ void make_dft(int lane, v2f dr[4], v2f dsn[4]) {
  const int lo  = lane & 15;
  const int hi2 = (lane >> 4) << 1;
#pragma unroll
  for (int c = 0; c < 4; ++c) {
    int k0 = 4 * c + hi2;
    float a0 = (float)((lo * k0) & 15) * (6.28318530718f / 16.f);
    float a1 = (float)((lo * (k0 + 1)) & 15) * (6.28318530718f / 16.f);
    dr[c].x  = __cosf(a0); dr[c].y  = __cosf(a1);
    dsn[c].x = __sinf(a0); dsn[c].y = __sinf(a1);
  }
}

// One 256-point FFT per wave32.  DIR=-1 forward, +1 inverse (unnormalized).
// Input: B-layout chunks bxr/bxi (bxi ignored if !CIN).
// Output: D-layout accumulators oR/oI: lane l, slot r holds X[k1 + 16*k2]
// with k1 = r + 8*(l>=16), k2 = l&15.  (oI skipped if !COUT.)
// lr/li: per-wave 256-float LDS scratch each (mid-FFT transpose).
template <int DIR, bool CIN, bool COUT>
__device__ __forceinline__ void fft256(const v2f *bxr, const v2f *bxi,
                                       const v2f *dr, const v2f *dsn,
                                       float *lr, float *li, int lane,
                                       v8f &oR, v8f &oI) {
  const int lo  = lane & 15;
  const int hi2 = (lane >> 4) << 1;
  const int hh  = (lane >> 4) << 3;

  // Stage 1: A1 = D16 * Xmat   (complex)
  v8f cr = {}, ci = {};
#pragma unroll
  for (int c = 0; c < 4; ++c) {
    v2f di = (DIR < 0) ? -dsn[c] : dsn[c];
    v2f nd = -di;
    cr = wmma4(dr[c], bxr[c], cr);
    ci = wmma4(di,    bxr[c], ci);
    if (CIN) {
      cr = wmma4(nd,    bxi[c], cr);
      ci = wmma4(dr[c], bxi[c], ci);
    }
  }

  // Twiddle (w256^{DIR*k1*n2}) and transpose via per-wave LDS tile.
#pragma unroll
  for (int r = 0; r < 8; ++r) {
    int m = r + hh;                     // k1
    float ang = (float)((m * lo) & 255) * (6.28318530718f / 256.f);
    float tc = __cosf(ang);
    float ts = __sinf(ang) * (float)DIR;
    lr[m * 16 + lo] = cr[r] * tc - ci[r] * ts;
    li[m * 16 + lo] = cr[r] * ts + ci[r] * tc;
  }
  asm volatile("s_wait_dscnt 0" ::: "memory");

  // Stage 2: X = A1tw * D16   (A-operand read transposed from LDS)
  v8f xr = {}, xi = {};
#pragma unroll
  for (int c = 0; c < 4; ++c) {
    int k0 = 4 * c + hi2;
    v2f ar, ai;
    ar.x = lr[lo * 16 + k0]; ar.y = lr[lo * 16 + k0 + 1];
    ai.x = li[lo * 16 + k0]; ai.y = li[lo * 16 + k0 + 1];
    v2f di = (DIR < 0) ? -dsn[c] : dsn[c];
    v2f nd = -di;
    xr = wmma4(ar, dr[c], xr);
    xr = wmma4(ai, nd,    xr);
    if (COUT) {
      xi = wmma4(ar, di,    xi);
      xi = wmma4(ai, dr[c], xi);
    }
  }
  oR = xr;
  if (COUT) oI = xi;
}

// ---------------- kernel 1: forward row FFT (real -> complex) --------------
__global__ void k_rowfft_fwd(const float *__restrict__ x,
                             float *__restrict__ Xre, float *__restrict__ Xim) {
  __shared__ float lds[8][2][256];
  const int lane = threadIdx.x & 31, wave = threadIdx.x >> 5;
  const int row  = blockIdx.x * 8 + wave;        // 0..24575
  const int lo   = lane & 15;
  const int hi2  = (lane >> 4) << 1, hh = (lane >> 4) << 3;
  const float *src = x + (size_t)row * 256;

  v2f dr[4], dsn[4];
  make_dft(lane, dr, dsn);

  v2f bxr[4], bxi[4];
#pragma unroll
  for (int c = 0; c < 4; ++c) {
    int n0 = 16 * (4 * c + hi2) + lo;
    bxr[c].x = src[n0];
    bxr[c].y = src[n0 + 16];
    bxi[c]   = v2f{0.f, 0.f};
  }
  v8f oR, oI;
  fft256<-1, false, true>(bxr, bxi, dr, dsn,
                          &lds[wave][0][0], &lds[wave][1][0], lane, oR, oI);
  float *dre = Xre + (size_t)row * 256;
  float *dim = Xim + (size_t)row * 256;
#pragma unroll
  for (int r = 0; r < 8; ++r) {
    int k = (r + hh) + 16 * lo;
    dre[k] = oR[r];
    dim[k] = oI[r];
  }
}

// -------- kernel 2: col FFT -> mask/256 * X - s -> inverse col FFT ---------
__global__ void k_colfft_mask(float *__restrict__ Xre, float *__restrict__ Xim,
                              const float *__restrict__ mask,
                              const float *__restrict__ sre,
                              const float *__restrict__ sim) {
  __shared__ float lds[8][2][256];
  const int lane = threadIdx.x & 31, wave = threadIdx.x >> 5;
  const int col  = blockIdx.x * 8 + wave;        // 0..24575
  const int img  = col >> 8, w = col & 255;
  const size_t base = (size_t)img * HW + w;
  const int lo = lane & 15;
  const int hi2 = (lane >> 4) << 1, hh = (lane >> 4) << 3;

  v2f dr[4], dsn[4];
  make_dft(lane, dr, dsn);

  v2f bxr[4], bxi[4];
#pragma unroll
  for (int c = 0; c < 4; ++c) {
    int n0 = 16 * (4 * c + hi2) + lo;
    bxr[c].x = Xre[base + (size_t)n0 * 256];
    bxr[c].y = Xre[base + (size_t)(n0 + 16) * 256];
    bxi[c].x = Xim[base + (size_t)n0 * 256];
    bxi[c].y = Xim[base + (size_t)(n0 + 16) * 256];
  }
  float *lr = &lds[wave][0][0], *li = &lds[wave][1][0];
  v8f oR, oI;
  fft256<-1, true, true>(bxr, bxi, dr, dsn, lr, li, lane, oR, oI);

  // frequency-domain: y = mask * (F x / 256) - s   (linear layout in LDS)
#pragma unroll
  for (int r = 0; r < 8; ++r) {
    int k = (r + hh) + 16 * lo;
    size_t off = base + (size_t)k * 256;
    float mv = mask[off] * (1.0f / 256.0f);
    lr[k] = oR[r] * mv - sre[off];
    li[k] = oI[r] * mv - sim[off];
  }
  asm volatile("s_wait_dscnt 0" ::: "memory");
#pragma unroll
  for (int c = 0; c < 4; ++c) {
    int n0 = 16 * (4 * c + hi2) + lo;
    bxr[c].x = lr[n0]; bxr[c].y = lr[n0 + 16];
    bxi[c].x = li[n0]; bxi[c].y = li[n0 + 16];
  }
  fft256<1, true, true>(bxr, bxi, dr, dsn, lr, li, lane, oR, oI);
#pragma unroll
  for (int r = 0; r < 8; ++r) {
    int k = (r + hh) + 16 * lo;
    Xre[base + (size_t)k * 256] = oR[r];
    Xim[base + (size_t)k * 256] = oI[r];
  }
}

// ------------- kernel 3: inverse row FFT, real part, /256 ------------------
__global__ void k_rowfft_inv(const float *__restrict__ Xre,
                             const float *__restrict__ Xim,
                             float *__restrict__ data) {
  __shared__ float lds[8][2][256];
  const int lane = threadIdx.x & 31, wave = threadIdx.x >> 5;
  const int row  = blockIdx.x * 8 + wave;
  const int lo   = lane & 15;
  const int hi2  = (lane >> 4) << 1, hh = (lane >> 4) << 3;
  const float *sre = Xre + (size_t)row * 256;
  const float *sim = Xim + (size_t)row * 256;

  v2f dr[4], dsn[4];
  make_dft(lane, dr, dsn);

  v2f bxr[4], bxi[4];
#pragma unroll
  for (int c = 0; c < 4; ++c) {
    int n0 = 16 * (4 * c + hi2) + lo;
    bxr[c].x = sre[n0]; bxr[c].y = sre[n0 + 16];
    bxi[c].x = sim[n0]; bxi[c].y = sim[n0 + 16];
  }
  v8f oR, oI;
  fft256<1, true, false>(bxr, bxi, dr, dsn,
                         &lds[wave][0][0], &lds[wave][1][0], lane, oR, oI);
  float *dst = data + (size_t)row * 256;
#pragma unroll
  for (int r = 0; r < 8; ++r) {
    int k = (r + hh) + 16 * lo;
    dst[k] = oR[r] * (1.0f / 256.0f);
  }
}

// ------- kernel 4: phi(b,f,h,w) from T-reduced conv magnitude --------------
__global__ void k_phi(const float *__restrict__ x, const float *__restrict__ filt,
                      const float *__restrict__ vals, float *__restrict__ phi) {
  __shared__ float xs[18][20];
  const int b = blockIdx.y;                 // 0..3
  const int tile = blockIdx.x;              // 0..255
  const int th = (tile >> 4) * 16, tw = (tile & 15) * 16;
  const int tx = threadIdx.x & 15, ty = threadIdx.x >> 4;

  float wgt[3][9];
#pragma unroll
  for (int f = 0; f < 3; ++f)
#pragma unroll
    for (int q = 0; q < 9; ++q) wgt[f][q] = filt[f * 9 + q];

  float acc[3] = {0.f, 0.f, 0.f};
  for (int t = 0; t < 24; ++t) {
    const float *img = x + (size_t)(b * 24 + t) * HW;
    __syncthreads();
    for (int i = threadIdx.x; i < 18 * 18; i += 256) {
      int iy = i / 18, ix = i - iy * 18;
      int gy = th + iy - 1, gx = tw + ix - 1;
      float v = 0.f;
      if (gy >= 0 && gy < 256 && gx >= 0 && gx < 256) v = img[gy * 256 + gx];
      xs[iy][ix] = v;
    }
    __syncthreads();
    float f0 = 0.f, f1 = 0.f, f2 = 0.f;
#pragma unroll
    for (int q = 0; q < 9; ++q) {
      int dy = q / 3, dx = q - dy * 3;
      float xv = xs[ty + dy][tx + dx];
      f0 += wgt[0][q] * xv;
      f1 += wgt[1][q] * xv;
      f2 += wgt[2][q] * xv;
    }
    acc[0] += f0 * f0; acc[1] += f1 * f1; acc[2] += f2 * f2;
  }
  const int h = th + ty, w = tw + tx;
#pragma unroll
  for (int f = 0; f < 3; ++f) {
    float mag = sqrtf(acc[f] * (1.0f / 24.0f) + 1e-8f);
    float z = fminf(mag, 1.5f);
    // GRID = linspace(0,1.5,100): searchsorted-left == ceil(z/step)
    int idx = (int)ceilf(z * (99.0f / 1.5f));
    idx = idx < 1 ? 1 : (idx > 99 ? 99 : idx);
    float g0 = (float)(idx - 1) * (1.5f / 99.0f);
    float y0 = vals[f * 100 + idx - 1];
    float y1 = vals[f * 100 + idx];
    phi[(((size_t)b * 3 + f) * 256 + h) * 256 + w] =
        y0 + (y1 - y0) * (99.0f / 1.5f) * (z - g0);
  }
}

// ------- kernel 5: fused  reg = W^T(phi .* Wx),  momentum update -----------
__global__ void k_update(float *__restrict__ x, float *__restrict__ m,
                         const float *__restrict__ data,
                         const float *__restrict__ phi,
                         const float *__restrict__ filt,
                         const float *__restrict__ alpha,
                         const float *__restrict__ mu, int layer) {
  __shared__ float xs[20][22];
  __shared__ float tp[3][18][20];
  const int bt = blockIdx.y;                // 0..95
  const int b = bt / 24;
  const int tile = blockIdx.x;
  const int th = (tile >> 4) * 16, tw = (tile & 15) * 16;
  const int tx = threadIdx.x & 15, ty = threadIdx.x >> 4;
  float *img = x + (size_t)bt * HW;

  float wgt[3][9];
#pragma unroll
  for (int f = 0; f < 3; ++f)
#pragma unroll
    for (int q = 0; q < 9; ++q) wgt[f][q] = filt[f * 9 + q];
  const float a = alpha[layer], muv = mu[layer];

  // x tile with halo 2
  for (int i = threadIdx.x; i < 20 * 20; i += 256) {
    int iy = i / 20, ix = i - iy * 20;
    int gy = th + iy - 2, gx = tw + ix - 2;
    xs[iy][ix] = (gy >= 0 && gy < 256 && gx >= 0 && gx < 256)
                     ? img[gy * 256 + gx] : 0.f;
  }
  __syncthreads();

  // tmp_f = phi_f * conv_f(x) on 18x18 ring (zero outside image: SAME pad)
  for (int i = threadIdx.x; i < 18 * 18; i += 256) {
    int iy = i / 18, ix = i - iy * 18;
    int gy = th + iy - 1, gx = tw + ix - 1;
    bool in = (gy >= 0 && gy < 256 && gx >= 0 && gx < 256);
    float f0 = 0.f, f1 = 0.f, f2 = 0.f;
#pragma unroll
    for (int q = 0; q < 9; ++q) {
      int dy = q / 3, dx = q - dy * 3;
      float xv = xs[iy + dy][ix + dx];
      f0 += wgt[0][q] * xv;
      f1 += wgt[1][q] * xv;
      f2 += wgt[2][q] * xv;
    }
    float p0 = 0.f, p1 = 0.f, p2 = 0.f;
    if (in) {
      size_t pb = ((size_t)b * 3) * HW + (size_t)gy * 256 + gx;
      p0 = phi[pb];
      p1 = phi[pb + HW];
      p2 = phi[pb + 2 * (size_t)HW];
    }
    tp[0][iy][ix] = in ? f0 * p0 : 0.f;
    tp[1][iy][ix] = in ? f1 * p1 : 0.f;
    tp[2][iy][ix] = in ? f2 * p2 : 0.f;
  }
  __syncthreads();

  // adjoint conv: out(p) = sum_f sum_q w[f][8-q] * tmp_f(p + q - 1)
  float reg = 0.f;
#pragma unroll
  for (int q = 0; q < 9; ++q) {
    int dy = q / 3, dx = q - dy * 3;
    reg += wgt[0][8 - q] * tp[0][ty + dy][tx + dx];
    reg += wgt[1][8 - q] * tp[1][ty + dy][tx + dx];
    reg += wgt[2][8 - q] * tp[2][ty + dy][tx + dx];
  }
  size_t off = (size_t)bt * HW + (size_t)(th + ty) * 256 + (tw + tx);
  float g = a * (data[off] + reg);
  float mn = muv * m[off] + g;
  m[off] = mn;
  x[off] = xs[ty + 2][tx + 2] - mn;
}

// ---------------- init: x = x0, m = 0 --------------------------------------
__global__ void k_init(const float *__restrict__ x0, float *__restrict__ x,
                       float *__restrict__ m, size_t n) {
  size_t i = (size_t)blockIdx.x * 256 + threadIdx.x;
  if (i < n) { x[i] = x0[i]; m[i] = 0.f; }
}

// ---------------------------------------------------------------------------
extern "C" void kernel_launch(void *const *d_in, const int *in_sizes, int n_in,
                              void *d_out, int out_size, void *d_ws,
                              size_t ws_size, hipStream_t stream) {
  (void)in_sizes; (void)n_in; (void)out_size; (void)ws_size;
  const float *x0    = (const float *)d_in[0];
  const float *sre   = (const float *)d_in[1];
  const float *sim   = (const float *)d_in[2];
  const float *mask  = (const float *)d_in[3];
  const float *alpha = (const float *)d_in[4];
  const float *mu    = (const float *)d_in[5];
  const float *filt  = (const float *)d_in[6];
  const float *actv  = (const float *)d_in[7];

  float *x = (float *)d_out;
  float *Xre  = (float *)d_ws;
  float *Xim  = Xre + NPIX;
  float *data = Xim + NPIX;
  float *mbuf = data + NPIX;
  float *phi  = mbuf + NPIX;      // 4*3*65536 floats

  k_init<<<(unsigned)(NPIX / 256), 256, 0, stream>>>(x0, x, mbuf, NPIX);

  for (int l = 0; l < 10; ++l) {
    k_rowfft_fwd<<<3072, 256, 0, stream>>>(x, Xre, Xim);
    k_colfft_mask<<<3072, 256, 0, stream>>>(Xre, Xim, mask, sre, sim);
    k_rowfft_inv<<<3072, 256, 0, stream>>>(Xre, Xim, data);
    k_phi<<<dim3(256, 4), 256, 0, stream>>>(x, filt + l * 27, actv + l * 300, phi);
    k_update<<<dim3(256, 96), 256, 0, stream>>>(x, mbuf, data, phi,
                                                filt + l * 27, alpha, mu, l);
  }
}